// fccaps_18562848653793
// MI455X (gfx1250) — compile-verified
//
#include <hip/hip_runtime.h>
#include <math.h>

typedef float v2f __attribute__((ext_vector_type(2)));
typedef float v8f __attribute__((ext_vector_type(8)));

#define B_TOT 128
#define N_IN  1152
#define D_IN  8
#define N_OUT 32
#define D_OUT 16
#define I_PER_BLOCK 72   // 1152 / 16 chunks
#define EPS_SQ 1e-20f

// DPP16 ROW_XMASK butterfly (within 16-lane rows). Pure VALU.
#define DPP_XADD(x, ctrl)                                                     \
    ((x) + __int_as_float(__builtin_amdgcn_update_dpp(                        \
               0, __float_as_int(x), (ctrl), 0xF, 0xF, true)))
#define DPP_XMAX(x, ctrl)                                                     \
    fmaxf((x), __int_as_float(__builtin_amdgcn_update_dpp(                    \
                   0, __float_as_int(x), (ctrl), 0xF, 0xF, true)))

// ---------------------------------------------------------------- zero s
__global__ __launch_bounds__(256) void caps_zero(float* __restrict__ s) {
    int idx = blockIdx.x * 256 + threadIdx.x;
    if (idx < B_TOT * N_OUT * D_OUT) s[idx] = 0.0f;
}

// ---------------------------------------------------------------- fused routing
// grid.x = 8 (batch groups of 16), grid.y = 16 (i chunks of 72), block = 256 (8 waves)
__global__ __launch_bounds__(256) void caps_route(const float* __restrict__ u,
                                                  const float* __restrict__ W,
                                                  const float* __restrict__ bias,
                                                  float* __restrict__ s_out) {
    const int lane = threadIdx.x & 31;
    const int wv   = threadIdx.x >> 5;    // wave 0..7 -> owns j = wv*4 .. wv*4+3
    const int half = lane >> 4;           // 0/1 (lane group)
    const int lm   = lane & 15;           // 0..15
    const int b_base  = blockIdx.x * 16;
    const int i_begin = blockIdx.y * I_PER_BLOCK;

    __shared__ float a_lds[16 * N_OUT];   // agreement scores a[b_local, j]
    __shared__ float c_lds[16 * N_OUT];   // routing coeffs   c[b_local, j]

    v8f sacc[4] = {};                     // s partial: 4 j-tiles x (16b x 16m)

    // u rows for this lane's batch (A-matrix M = lm)
    const float* up_base = u + (size_t)(b_base + lm) * (N_IN * D_IN);

    for (int ii = 0; ii < I_PER_BLOCK; ++ii) {
        const int i = i_begin + ii;

        // ---- A fragments: 16x4 fp32 layout: lanes 0-15 hold K={v,v+1}@half0,
        //      lanes 16-31 K shifted by 2.  Two fragments cover K=0..7.
        const float* up = up_base + (size_t)i * D_IN + 2 * half;
        v2f a0, a1;
        a0[0] = up[0]; a0[1] = up[1];     // K = 2*half + {0,1}
        a1[0] = up[4]; a1[1] = up[5];     // K = 4 + 2*half + {0,1}

        // ---- B fragments + WMMA: W[i, j, k, m], B tile is 4x16 (KxN=m)
        const float* wp = W + (size_t)i * (N_OUT * D_IN * D_OUT)
                            + (size_t)(wv * 4) * (D_IN * D_OUT)
                            + 2 * half * D_OUT + lm;
        // speculative prefetch of next i's W slab for this wave
        __builtin_prefetch((const void*)(wp + N_OUT * D_IN * D_OUT), 0, 0);

        v8f uh[4];
        #pragma unroll
        for (int t = 0; t < 4; ++t) {
            const float* wjt = wp + t * (D_IN * D_OUT);
            v2f b0, b1;
            b0[0] = wjt[0];   b0[1] = wjt[16];    // K = 2*half + {0,1}
            b1[0] = wjt[64];  b1[1] = wjt[80];    // K = 4 + 2*half + {0,1}
            v8f c = {};
            c = __builtin_amdgcn_wmma_f32_16x16x4_f32(false, a0, false, b0,
                                                      (short)0, c, false, false);
            c = __builtin_amdgcn_wmma_f32_16x16x4_f32(false, a1, false, b1,
                                                      (short)0, c, false, false);
            uh[t] = c;   // u_hat tile: rows=16 batches, cols=16 m, for j=wv*4+t
        }

        // ---- agreement a[b,j] = sum_m uhat^2 / sqrt(16)   (DPP row_xmask butterfly)
        #pragma unroll
        for (int t = 0; t < 4; ++t) {
            const int j = wv * 4 + t;
            float asum[8];
            #pragma unroll
            for (int r = 0; r < 8; ++r) {
                float x  = uh[t][r];
                float sq = x * x;
                sq = DPP_XADD(sq, 0x161);   // xor 1
                sq = DPP_XADD(sq, 0x162);   // xor 2
                sq = DPP_XADD(sq, 0x164);   // xor 4
                sq = DPP_XADD(sq, 0x168);   // xor 8 -> full 16-lane row sum
                asum[r] = sq;
            }
            if (lm == 0) {                  // one writer per 16-lane half
                #pragma unroll
                for (int r = 0; r < 8; ++r)
                    a_lds[(r + 8 * half) * N_OUT + j] = asum[r] * 0.25f;
            }
        }
        __syncthreads();

        // ---- softmax over j (+ learned bias): wave-parallel, branch-free.
        //      Wave wv handles local batches 2*wv and 2*wv+1; lane = j.
        {
            const float bs = bias[i * N_OUT + lane];
            #pragma unroll
            for (int p = 0; p < 2; ++p) {
                const int bb = 2 * wv + p;
                const float av = a_lds[bb * N_OUT + lane];
                float mx = av;                       // wave-wide max over 32 j
                mx = DPP_XMAX(mx, 0x161);
                mx = DPP_XMAX(mx, 0x162);
                mx = DPP_XMAX(mx, 0x164);
                mx = DPP_XMAX(mx, 0x168);
                mx = fmaxf(mx, __shfl_xor(mx, 16, 32));
                const float e = __expf(av - mx);
                float sm = e;                        // wave-wide sum over 32 j
                sm = DPP_XADD(sm, 0x161);
                sm = DPP_XADD(sm, 0x162);
                sm = DPP_XADD(sm, 0x164);
                sm = DPP_XADD(sm, 0x168);
                sm += __shfl_xor(sm, 16, 32);
                c_lds[bb * N_OUT + lane] = e / sm + bs;
            }
        }
        __syncthreads();

        // ---- s[b,j,m] += c[b,j] * u_hat[b,j,m]
        //      c for this wave's 4 j's are contiguous -> one ds_load_b128 per row
        const float4* c4 = reinterpret_cast<const float4*>(c_lds);
        #pragma unroll
        for (int r = 0; r < 8; ++r) {
            float4 cc = c4[(r + 8 * half) * (N_OUT / 4) + wv];
            sacc[0][r] += cc.x * uh[0][r];
            sacc[1][r] += cc.y * uh[1][r];
            sacc[2][r] += cc.z * uh[2][r];
            sacc[3][r] += cc.w * uh[3][r];
        }
        // barrier 2 above orders a_lds(i+1) writes after softmax reads of a_lds(i);
        // barrier 1 orders c_lds(i+1) writes after these c_lds(i) reads.
    }

    // ---- cross-i-chunk reduction into global s
    #pragma unroll
    for (int t = 0; t < 4; ++t) {
        const int j = wv * 4 + t;
        #pragma unroll
        for (int r = 0; r < 8; ++r) {
            const int b_ = b_base + r + 8 * half;
            atomicAdd(&s_out[((size_t)b_ * N_OUT + j) * D_OUT + lm], sacc[t][r]);
        }
    }
}

// ---------------------------------------------------------------- squash (in place)
__global__ __launch_bounds__(256) void caps_squash(float* __restrict__ s) {
    int idx = blockIdx.x * 256 + threadIdx.x;     // one thread per (b, j)
    if (idx >= B_TOT * N_OUT) return;
    float* p = s + (size_t)idx * D_OUT;
    float v[D_OUT];
    float n2 = 0.0f;
    #pragma unroll
    for (int m = 0; m < D_OUT; ++m) { v[m] = p[m]; n2 += v[m] * v[m]; }
    const float n = sqrtf(n2);
    const float scale = n2 / (1.0f + n2) / (n + EPS_SQ);
    #pragma unroll
    for (int m = 0; m < D_OUT; ++m) p[m] = v[m] * scale;
}

// ---------------------------------------------------------------- launcher
extern "C" void kernel_launch(void* const* d_in, const int* in_sizes, int n_in,
                              void* d_out, int out_size, void* d_ws, size_t ws_size,
                              hipStream_t stream) {
    const float* u    = (const float*)d_in[0];   // [128, 1152, 8]
    const float* W    = (const float*)d_in[1];   // [1152, 32, 8, 16]
    const float* bias = (const float*)d_in[2];   // [1152, 32]
    float* out = (float*)d_out;                  // [128, 32, 16]

    caps_zero<<<(B_TOT * N_OUT * D_OUT + 255) / 256, 256, 0, stream>>>(out);

    dim3 grid(B_TOT / 16, N_IN / I_PER_BLOCK);   // 8 x 16 = 128 blocks
    caps_route<<<grid, 256, 0, stream>>>(u, W, bias, out);

    caps_squash<<<(B_TOT * N_OUT + 255) / 256, 256, 0, stream>>>(out);
}